// AFM_42984032698906
// MI455X (gfx1250) — compile-verified
//
#include <hip/hip_runtime.h>
#include <hip/hip_bf16.h>

typedef __attribute__((ext_vector_type(2))) float v2f;
typedef __attribute__((ext_vector_type(8))) float v8f;

#define LDST 68  // LDS row stride: multiple of 4 (float4 align), 68%64=4 avoids column bank conflicts

__global__ __launch_bounds__(256) void afm_kernel(
    const float* __restrict__ emb,   // [256,64,64]
    const float* __restrict__ wW,    // [64,64]  (e,d)
    const float* __restrict__ wB,    // [64]
    const float* __restrict__ aW,    // [64]
    const float* __restrict__ aB,    // [1]
    float* __restrict__ out)         // alphas [256*64*64] ++ value [256*64*64*64]
{
    __shared__ float sE [64 * LDST];   // E_b, row j, col d
    __shared__ float sWT[64 * LDST];   // W transposed: sWT[d][e] = wW[e*64+d]
    __shared__ float sWB[64];
    __shared__ float sAW[64];
    __shared__ float sLogit[64];

    const int b    = blockIdx.x;
    const int tid  = threadIdx.x;
    const int wave = tid >> 5;
    const int lane = tid & 31;
    const int m    = lane & 15;   // position within 16-lane half
    const int h    = lane >> 4;   // which half of the wave

    // ---- stage E_b into LDS (coalesced float4 loads) ----
    const float* Eb = emb + b * 4096;
    #pragma unroll
    for (int c = 0; c < 4; ++c) {
        int idx4 = tid + c * 256;
        float4 v = ((const float4*)Eb)[idx4];
        int el = idx4 * 4;
        *(float4*)&sE[(el >> 6) * LDST + (el & 63)] = v;
    }
    // ---- stage W^T into LDS (coalesced global reads, transposed LDS writes) ----
    #pragma unroll
    for (int c = 0; c < 16; ++c) {
        int idx = tid + c * 256;
        sWT[(idx & 63) * LDST + (idx >> 6)] = wW[idx];
    }
    if (tid < 64) { sWB[tid] = wB[tid]; sAW[tid] = aW[tid]; }
    const float ab = aB[0];
    __syncthreads();

    // tile assignment: wave w -> output tiles (tm, tn0) and (tm, tn0+1)
    const int tm  = wave >> 1;
    const int tn0 = (wave & 1) * 2;
    const int jrowA = tm * 16 + m;          // A-matrix row this lane supplies

    // ---- hoist i-invariant fragments into registers ----
    // B fragments (W^T) never change; raw A-row values only get scaled by emb_i.
    v2f Bf0[16], Bf1[16], Ar[16];
    #pragma unroll
    for (int ks = 0; ks < 16; ++ks) {
        int k0 = ks * 4 + 2 * h;            // lanes 0-15: K=4ks,4ks+1 ; lanes 16-31: +2
        Bf0[ks].x = sWT[(k0)     * LDST + tn0 * 16 + m];
        Bf0[ks].y = sWT[(k0 + 1) * LDST + tn0 * 16 + m];
        Bf1[ks].x = sWT[(k0)     * LDST + (tn0 + 1) * 16 + m];
        Bf1[ks].y = sWT[(k0 + 1) * LDST + (tn0 + 1) * 16 + m];
        Ar[ks].x  = sE [(jrowA)  * LDST + k0];
        Ar[ks].y  = sE [(jrowA)  * LDST + k0 + 1];
    }

    const int       alphaBase = b * 64 * 64;
    const long long valueBase = 1048576LL + (long long)b * 64 * 64 * 64;

    for (int i = 0; i < 64; ++i) {
        __syncthreads();                     // protect sLogit from prior softmax readers
        if (tid < 64) sLogit[tid] = ab;

        const float* ei = &sE[i * LDST];

        // ---- value[b,i,j,:] = E_b[j,:] * E_b[i,:]  (coalesced float4 stores) ----
        {
            int j = tid >> 2;
            int q = tid & 3;
            const float* ej = &sE[j * LDST];
            float* vout = out + valueBase + (long long)(i * 64 + j) * 64;
            #pragma unroll
            for (int c = 0; c < 4; ++c) {
                int d = q * 4 + c * 16;
                float4 v;
                v.x = ej[d + 0] * ei[d + 0];
                v.y = ej[d + 1] * ei[d + 1];
                v.z = ej[d + 2] * ei[d + 2];
                v.w = ej[d + 3] * ei[d + 3];
                *(float4*)(vout + d) = v;
            }
        }
        __syncthreads();                     // sLogit init visible before atomics

        // ---- QK = (E_b * diag(E_i)) @ W^T : two 16x16 tiles/wave, K=64 in 16 steps ----
        v8f acc0 = {};
        v8f acc1 = {};
        #pragma unroll
        for (int ks = 0; ks < 16; ++ks) {
            int k0 = ks * 4 + 2 * h;
            v2f a;
            a.x = Ar[ks].x * ei[k0];         // broadcast LDS read of emb_i
            a.y = Ar[ks].y * ei[k0 + 1];
            acc0 = __builtin_amdgcn_wmma_f32_16x16x4_f32(false, a, false, Bf0[ks],
                                                         (short)0, acc0, false, false);
            acc1 = __builtin_amdgcn_wmma_f32_16x16x4_f32(false, a, false, Bf1[ks],
                                                         (short)0, acc1, false, false);
        }

        // ---- bias + LeakyReLU + dot(a_W), reduce over e, accumulate logits ----
        #pragma unroll
        for (int t = 0; t < 2; ++t) {
            v8f acc = t ? acc1 : acc0;
            int e = (tn0 + t) * 16 + m;
            float bias = sWB[e];
            float aw   = sAW[e];
            #pragma unroll
            for (int r = 0; r < 8; ++r) {
                float qv = acc[r] + bias;
                qv = (qv >= 0.f) ? qv : 0.01f * qv;   // LeakyReLU
                float p = qv * aw;
                p += __shfl_xor(p, 1, 32);
                p += __shfl_xor(p, 2, 32);
                p += __shfl_xor(p, 4, 32);
                p += __shfl_xor(p, 8, 32);            // sum over 16 N-lanes
                if (m == 0) {
                    int j = tm * 16 + r + 8 * h;      // C/D layout: VGPR r -> rows r, r+8
                    atomicAdd(&sLogit[j], p);         // ds_add_f32
                }
            }
        }
        __syncthreads();

        // ---- softmax over j (row length 64), threads 0..63 ----
        if (tid < 64) {
            float mx = -3.402823466e38f;
            #pragma unroll
            for (int j = 0; j < 64; ++j) mx = fmaxf(mx, sLogit[j]);
            float s = 0.f;
            #pragma unroll
            for (int j = 0; j < 64; ++j) s += __expf(sLogit[j] - mx);
            out[alphaBase + i * 64 + tid] = __expf(sLogit[tid] - mx) / s;
        }
    }
}

extern "C" void kernel_launch(void* const* d_in, const int* in_sizes, int n_in,
                              void* d_out, int out_size, void* d_ws, size_t ws_size,
                              hipStream_t stream) {
    const float* emb = (const float*)d_in[0];
    const float* wW  = (const float*)d_in[1];
    const float* wB  = (const float*)d_in[2];
    const float* aW  = (const float*)d_in[3];
    const float* aB  = (const float*)d_in[4];
    float* out = (float*)d_out;
    afm_kernel<<<256, 256, 0, stream>>>(emb, wW, wB, aW, aB, out);
}